// GENEncoder_1640677507754
// MI455X (gfx1250) — compile-verified
//
#include <hip/hip_runtime.h>
#include <math.h>
#include <stdint.h>

typedef __attribute__((ext_vector_type(2))) float v2f;
typedef __attribute__((ext_vector_type(4))) float v4f;
typedef __attribute__((ext_vector_type(8))) float v8f;

namespace {

constexpr int Nn = 20000, Ee = 320000, Gg = 64;
constexpr int Ff = 128, EDd = 32, Hh = 300, HIDh = 600, DLl = 256, DOUTd = 128;
constexpr float BN_INV_F = 0.9999950000374997f;   // 1/sqrt(1+1e-5)
constexpr float EPS_F = 1e-7f;

enum { EPI_NONE = 0, EPI_BIAS = 1, EPI_BIAS_RELU = 2, EPI_RELU_BN = 3,
       EPI_EDGE_MAX = 4, EPI_EDGE_SOFT = 5 };

// Block tile: 32(M) x 64(N), K in chunks of 32, double-buffered LDS.
// 8 wave32s, each owns one 16x16 C tile (2x4 wave grid).
constexpr int BM = 32, BN = 64, BK = 32;
constexpr int LDAS = BK + 4;  // 36 floats: row stride 144B (16B aligned, c%4==0)
constexpr int LDBS = BN + 4;  // 68 floats: row stride 272B (16B aligned, c%4==0)

// D = A[MxK] @ B[KxN] + fused epilogue. Exact f32 via V_WMMA_F32_16X16X4_F32.
// Fragment layouts per CDNA5 ISA 7.12.2:
//   A 16x4 : lane l -> row M = l&15, VGPR{0,1} = K = 2*(l>>4)+{0,1}
//   B 4x16 : lane l -> col N = l&15, VGPR{0,1} = K = 2*(l>>4)+{0,1}
//   C 16x16: VGPR r, lane l -> row M = r + 8*(l>>4), col N = l&15
// All K/N leading dims here are multiples of 4 -> float4/b128 staging is 16B
// aligned and never partially straddles an edge.
// Interior tiles are staged with GLOBAL_LOAD_ASYNC_TO_LDS_B128 (ASYNCcnt):
// the copy lands in LDS while the WMMA chain runs; boundary tiles fall back
// to guarded register staging (async loads cannot zero-fill OOB lanes).
template <int EPI>
__global__ __launch_bounds__(256) void wmma_gemm_kernel(
    const float* __restrict__ A, const float* __restrict__ B,
    float* __restrict__ C, int M, int N, int K,
    const float* __restrict__ bias, const float* __restrict__ gamma,
    const float* __restrict__ beta,
    const float* __restrict__ hsrc,      // [*,N] gather table (edge modes)
    const int* __restrict__ srcIdx, const int* __restrict__ dstIdx,
    float* __restrict__ segMax,          // EDGE_MAX: out (uint-max), EDGE_SOFT: in
    float* __restrict__ denom, float* __restrict__ numer) {
  __shared__ __align__(16) float As[2][BM * LDAS];
  __shared__ __align__(16) float Bs[2][BK * LDBS];

  const int t = threadIdx.x;
  const int lane = t & 31;
  const int wid = t >> 5;
  const int wm = wid >> 2;     // 0..1
  const int wn = wid & 3;      // 0..3
  const int lm = lane & 15;
  const int lh = lane >> 4;

  const int m0 = blockIdx.x * BM;
  const int n0 = blockIdx.y * BN;

  // Per-thread staging coordinates (one A float4, two B float4 per thread).
  const int ar = t >> 3;          // A tile row 0..31
  const int ac = (t & 7) * 4;     // A tile col {0,4,...,28}

  v8f acc = {};
  v4f aStage;
  v4f bStage[2];

  const bool mnInterior = (m0 + BM <= M) && (n0 + BN <= N);

  // ---- async fast path: global -> LDS DMA, no VGPR round trip ----
  auto stageAsync = [&](int k0, int pbuf) {
    {
      unsigned lofs = (unsigned)(uintptr_t)&As[pbuf][ar * LDAS + ac];
      const float* g = &A[(size_t)(m0 + ar) * K + (k0 + ac)];
      asm volatile("global_load_async_to_lds_b128 %0, %1, off"
                   :: "v"(lofs), "v"(g) : "memory");
    }
#pragma unroll
    for (int i = 0; i < 2; ++i) {
      int lin = t + i * 256;
      int r = lin >> 4, c = (lin & 15) * 4;
      unsigned lofs = (unsigned)(uintptr_t)&Bs[pbuf][r * LDBS + c];
      const float* g = &B[(size_t)(k0 + r) * N + (n0 + c)];
      asm volatile("global_load_async_to_lds_b128 %0, %1, off"
                   :: "v"(lofs), "v"(g) : "memory");
    }
  };

  // ---- boundary fallback: guarded register staging ----
  auto loadA = [&](int k0) {
    int gr = m0 + ar, gc = k0 + ac;
    v4f v = {};
    if (gr < M && gc + 3 < K) {
      v = *(const v4f*)&A[(size_t)gr * K + gc];
      int pc = gc + BK; pc = pc > K - 4 ? K - 4 : pc;
      __builtin_prefetch(&A[(size_t)gr * K + pc], 0, 1);  // global_prefetch_b8
    }
    aStage = v;
  };
  auto loadB = [&](int k0) {
#pragma unroll
    for (int i = 0; i < 2; ++i) {
      int lin = t + i * 256;
      int r = lin >> 4, c = (lin & 15) * 4;
      int gr = k0 + r, gc = n0 + c;
      v4f v = {};
      if (gr < K && gc + 3 < N) {
        v = *(const v4f*)&B[(size_t)gr * N + gc];
        int pr = gr + BK; pr = pr > K - 1 ? K - 1 : pr;
        __builtin_prefetch(&B[(size_t)pr * N + gc], 0, 1);
      }
      bStage[i] = v;
    }
  };
  auto storeA = [&](int p) { *(v4f*)&As[p][ar * LDAS + ac] = aStage; };
  auto storeB = [&](int p) {
#pragma unroll
    for (int i = 0; i < 2; ++i) {
      int lin = t + i * 256;
      int r = lin >> 4, c = (lin & 15) * 4;
      *(v4f*)&Bs[p][r * LDBS + c] = bStage[i];
    }
  };

  // Prologue: stage chunk 0 into buffer 0.
  const bool interior0 = mnInterior && (BK <= K);
  if (interior0) {
    stageAsync(0, 0);
  } else {
    loadA(0); loadB(0);
    storeA(0); storeB(0);
  }

  int p = 0;
  const int arow = (wm * 16 + lm) * LDAS;
  const int bcol = wn * 16 + lm;
  for (int k0 = 0; k0 < K; k0 += BK) {
    // Publish this chunk's staging: wait for our own async LDS writes, then
    // barrier so every wave's writes are visible (__syncthreads only covers
    // DScnt, not ASYNCcnt).
    asm volatile("s_wait_asynccnt 0x0" ::: "memory");
    __syncthreads();

    // Preload ALL fragments for this chunk so the 8 WMMAs issue as a pure
    // back-to-back accumulator chain.
    v2f af[BK / 4], bf[BK / 4];
    const float* Ap = As[p];
    const float* Bp = Bs[p];
#pragma unroll
    for (int q = 0; q < BK / 4; ++q) {
      const int ka = q * 4 + 2 * lh;            // even -> 8B-aligned v2f read
      af[q] = *(const v2f*)&Ap[arow + ka];
      bf[q].x = Bp[ka * LDBS + bcol];
      bf[q].y = Bp[(ka + 1) * LDBS + bcol];
    }

    const bool hasNext = (k0 + BK) < K;                    // scalar, uniform
    const bool nextInterior = hasNext && mnInterior && (k0 + 2 * BK <= K);
    if (hasNext) {
      if (nextInterior) stageAsync(k0 + BK, p ^ 1);        // DMA during WMMA
      else { loadA(k0 + BK); loadB(k0 + BK); }
    }

#pragma unroll
    for (int q = 0; q < BK / 4; ++q)
      acc = __builtin_amdgcn_wmma_f32_16x16x4_f32(
          /*neg_a=*/false, af[q], /*neg_b=*/false, bf[q],
          /*c_mod=*/(short)0, acc, /*reuse_a=*/false, /*reuse_b=*/false);

    if (hasNext) {
      if (!nextInterior) { storeA(p ^ 1); storeB(p ^ 1); }
      p ^= 1;
    }
  }

  const int colg = n0 + wn * 16 + lm;
#pragma unroll
  for (int r = 0; r < 8; ++r) {
    const int rowg = m0 + wm * 16 + r + 8 * lh;
    if (rowg >= M || colg >= N) continue;
    float v = acc[r];
    if constexpr (EPI == EPI_NONE) {
      C[(size_t)rowg * N + colg] = v;
    } else if constexpr (EPI == EPI_BIAS) {
      C[(size_t)rowg * N + colg] = v + bias[colg];
    } else if constexpr (EPI == EPI_BIAS_RELU) {
      C[(size_t)rowg * N + colg] = fmaxf(v + bias[colg], 0.f);
    } else if constexpr (EPI == EPI_RELU_BN) {
      C[(size_t)rowg * N + colg] =
          fmaxf(v + bias[colg], 0.f) * BN_INV_F * gamma[colg] + beta[colg];
    } else if constexpr (EPI == EPI_EDGE_MAX) {
      const int s = srcIdx[rowg], d = dstIdx[rowg];
      float msg = fmaxf(v + hsrc[(size_t)s * N + colg], 0.f) + EPS_F;
      // msg > 0 always => uint bit-pattern max == float max (identity 0.0).
      atomicMax(reinterpret_cast<unsigned int*>(segMax) + (size_t)d * N + colg,
                __float_as_uint(msg));
    } else if constexpr (EPI == EPI_EDGE_SOFT) {
      const int s = srcIdx[rowg], d = dstIdx[rowg];
      float msg = fmaxf(v + hsrc[(size_t)s * N + colg], 0.f) + EPS_F;
      float mx = segMax[(size_t)d * N + colg];
      float ex = expf(msg - mx);
      atomicAdd(&denom[(size_t)d * N + colg], ex);        // L2-resident (24MB << 192MB)
      atomicAdd(&numer[(size_t)d * N + colg], msg * ex);
    }
  }
}

__global__ void zero_kernel(float* __restrict__ p, long n) {
  long i = (long)blockIdx.x * blockDim.x + threadIdx.x;
  long stride = (long)gridDim.x * blockDim.x;
  for (; i < n; i += stride) p[i] = 0.f;
}

// out = xd + (denom>0 ? num/denom : 0), in place in xd.
__global__ void node_combine_kernel(float* __restrict__ xd,
                                    const float* __restrict__ num,
                                    const float* __restrict__ den, long n) {
  long i = (long)blockIdx.x * blockDim.x + threadIdx.x;
  long stride = (long)gridDim.x * blockDim.x;
  for (; i < n; i += stride) {
    float d = den[i];
    xd[i] = xd[i] + (d > 0.f ? num[i] / d : 0.f);
  }
}

__global__ void pool_count_kernel(const int* __restrict__ batch,
                                  float* __restrict__ cnt, int n) {
  int i = blockIdx.x * blockDim.x + threadIdx.x;
  if (i < n) atomicAdd(&cnt[batch[i]], 1.f);
}

__global__ void pool_sum_kernel(const float* __restrict__ fin,
                                const int* __restrict__ batch,
                                float* __restrict__ pooled, long n) {
  long i = (long)blockIdx.x * blockDim.x + threadIdx.x;
  if (i >= n) return;
  int node = (int)(i / DOUTd);
  int c = (int)(i % DOUTd);
  atomicAdd(&pooled[(size_t)batch[node] * DOUTd + c], fin[i]);
}

__global__ void pool_final_kernel(const float* __restrict__ pooled,
                                  const float* __restrict__ cnt,
                                  float* __restrict__ out, int n) {
  int i = blockIdx.x * blockDim.x + threadIdx.x;
  if (i >= n) return;
  int g = i / DOUTd;
  out[i] = pooled[i] / fmaxf(cnt[g], 1.f);
}

dim3 gemm_grid(int M, int N) {
  return dim3((unsigned)((M + BM - 1) / BM), (unsigned)((N + BN - 1) / BN));
}

}  // namespace

extern "C" void kernel_launch(void* const* d_in, const int* in_sizes, int n_in,
                              void* d_out, int out_size, void* d_ws, size_t ws_size,
                              hipStream_t stream) {
  (void)in_sizes; (void)n_in; (void)out_size; (void)ws_size;
  const float* x        = (const float*)d_in[0];
  const float* edgeAttr = (const float*)d_in[1];
  const int*   edgeIdx  = (const int*)d_in[2];
  const int*   batch    = (const int*)d_in[3];
  const float* W_src = (const float*)d_in[4];
  const float* W_edge= (const float*)d_in[5];
  const float* W_dst = (const float*)d_in[6];
  const float* W1 = (const float*)d_in[7];  const float* b1  = (const float*)d_in[8];
  const float* g1 = (const float*)d_in[9];  const float* be1 = (const float*)d_in[10];
  const float* W2 = (const float*)d_in[11]; const float* b2  = (const float*)d_in[12];
  const float* g2 = (const float*)d_in[13]; const float* be2 = (const float*)d_in[14];
  const float* W3 = (const float*)d_in[15]; const float* b3  = (const float*)d_in[16];
  const float* L1 = (const float*)d_in[17]; const float* bl1 = (const float*)d_in[18];
  const float* L2 = (const float*)d_in[19]; const float* bl2 = (const float*)d_in[20];

  const int* srcIdx = edgeIdx;        // edge_index[0,:]
  const int* dstIdx = edgeIdx + Ee;   // edge_index[1,:]

  float* ws = (float*)d_ws;
  const size_t S_NH = (size_t)Nn * Hh;
  float* buf_h   = ws;                              // [N,H]  h = x@W_src ; later out3
  float* buf_xd  = buf_h   + S_NH;                  // [N,H]  x@W_dst -> out (in place)
  float* buf_m   = buf_xd  + S_NH;                  // [N,H]  segment max ; later enc
  float* buf_den = buf_m   + S_NH;                  // [N,H]  softmax denom ; later fin
  float* buf_num = buf_den + S_NH;                  // [N,H]  softmax numerator
  float* buf_h1  = buf_num + S_NH;                  // [N,HID]
  float* buf_h2  = buf_h1  + (size_t)Nn * HIDh;     // [N,HID]
  float* pooled  = buf_h2  + (size_t)Nn * HIDh;     // [G,DOUT]
  float* cnt     = pooled  + (size_t)Gg * DOUTd;    // [G]
  float* out3 = buf_h;    // safe alias: h dead after edge passes
  float* enc  = buf_m;    // safe alias: m dead after edge-soft pass
  float* fin  = buf_den;  // safe alias: denom dead after node_combine

  dim3 blk(256);
  const float* np = nullptr; const int* ni = nullptr; float* nf = nullptr;

  // Zero reduction buffers (m, denom, num are contiguous) + pooled/cnt.
  zero_kernel<<<2048, 256, 0, stream>>>(buf_m, (long)(3 * S_NH));
  zero_kernel<<<64, 256, 0, stream>>>(pooled, (long)(Gg * DOUTd + Gg));

  // h = x @ W_src
  wmma_gemm_kernel<EPI_NONE><<<gemm_grid(Nn, Hh), blk, 0, stream>>>(
      x, W_src, buf_h, Nn, Hh, Ff, np, np, np, np, ni, ni, nf, nf, nf);
  // xd = x @ W_dst
  wmma_gemm_kernel<EPI_NONE><<<gemm_grid(Nn, Hh), blk, 0, stream>>>(
      x, W_dst, buf_xd, Nn, Hh, Ff, np, np, np, np, ni, ni, nf, nf, nf);

  // Edge pass 1: msg = relu(edge_attr@W_edge + h[src]) + eps ; m[dst] = max(msg)
  wmma_gemm_kernel<EPI_EDGE_MAX><<<gemm_grid(Ee, Hh), blk, 0, stream>>>(
      edgeAttr, W_edge, nf, Ee, Hh, EDd, np, np, np, buf_h, srcIdx, dstIdx,
      buf_m, nf, nf);
  // Edge pass 2 (recompute msg; cheaper than spilling 384MB): denom/num accumulation
  wmma_gemm_kernel<EPI_EDGE_SOFT><<<gemm_grid(Ee, Hh), blk, 0, stream>>>(
      edgeAttr, W_edge, nf, Ee, Hh, EDd, np, np, np, buf_h, srcIdx, dstIdx,
      buf_m, buf_den, buf_num);

  // out = xd + num/denom   (agg = sum(msg*ex)/sum(ex), per dst/channel)
  node_combine_kernel<<<4096, 256, 0, stream>>>(buf_xd, buf_num, buf_den, (long)S_NH);

  // MLP: h1 = relu(out@W1+b1)*BN_INV*g1 + be1
  wmma_gemm_kernel<EPI_RELU_BN><<<gemm_grid(Nn, HIDh), blk, 0, stream>>>(
      buf_xd, W1, buf_h1, Nn, HIDh, Hh, b1, g1, be1, np, ni, ni, nf, nf, nf);
  // h2 = relu(h1@W2+b2)*BN_INV*g2 + be2
  wmma_gemm_kernel<EPI_RELU_BN><<<gemm_grid(Nn, HIDh), blk, 0, stream>>>(
      buf_h1, W2, buf_h2, Nn, HIDh, HIDh, b2, g2, be2, np, ni, ni, nf, nf, nf);
  // out3 = h2@W3 + b3
  wmma_gemm_kernel<EPI_BIAS><<<gemm_grid(Nn, Hh), blk, 0, stream>>>(
      buf_h2, W3, out3, Nn, Hh, HIDh, b3, np, np, np, ni, ni, nf, nf, nf);
  // enc = relu(out3@L1 + bl1)
  wmma_gemm_kernel<EPI_BIAS_RELU><<<gemm_grid(Nn, DLl), blk, 0, stream>>>(
      out3, L1, enc, Nn, DLl, Hh, bl1, np, np, np, ni, ni, nf, nf, nf);
  // fin = enc@L2 + bl2
  wmma_gemm_kernel<EPI_BIAS><<<gemm_grid(Nn, DOUTd), blk, 0, stream>>>(
      enc, L2, fin, Nn, DOUTd, DLl, bl2, np, np, np, ni, ni, nf, nf, nf);

  // Global mean pool over graphs.
  pool_count_kernel<<<(Nn + 255) / 256, 256, 0, stream>>>(batch, cnt, Nn);
  pool_sum_kernel<<<(int)(((long)Nn * DOUTd + 255) / 256), 256, 0, stream>>>(
      fin, batch, pooled, (long)Nn * DOUTd);
  pool_final_kernel<<<(Gg * DOUTd + 255) / 256, 256, 0, stream>>>(
      pooled, cnt, (float*)d_out, Gg * DOUTd);
}